// GCN_Encoder_L1_18837726560469
// MI455X (gfx1250) — compile-verified
//
#include <hip/hip_runtime.h>
#include <hip/hip_bf16.h>

typedef float v2f __attribute__((ext_vector_type(2)));
typedef float v8f __attribute__((ext_vector_type(8)));

#define IN_C  256
#define OUT_C 64

__device__ __forceinline__ void atomAddF32(float* p, float v) {
  // Native global_atomic_add_f32, relaxed, device scope (no CAS loop).
  __hip_atomic_fetch_add(p, v, __ATOMIC_RELAXED, __HIP_MEMORY_SCOPE_AGENT);
}

// ---------------------------------------------------------------------------
// Workspace layout (floats):
//   [0, N)                 : deg, then overwritten in-place with dinv
//   [HOFF, HOFF + N*64)    : h = x @ W
// ---------------------------------------------------------------------------

// deg[i] = 1.0 (self-loop weight); also the re-init each call.
__global__ void k_init_deg(float* __restrict__ deg, int n) {
  int i = blockIdx.x * blockDim.x + threadIdx.x;
  if (i < n) deg[i] = 1.0f;
}

// deg[col[e]] += ew[e]
__global__ void k_deg(const int* __restrict__ ei, const float* __restrict__ ew,
                      float* __restrict__ deg, int n_edges) {
  int stride = gridDim.x * blockDim.x;
  for (int e = blockIdx.x * blockDim.x + threadIdx.x; e < n_edges; e += stride)
    atomAddF32(&deg[ei[n_edges + e]], ew[e]);
}

// deg -> dinv in place
__global__ void k_dinv(float* __restrict__ deg, int n) {
  int i = blockIdx.x * blockDim.x + threadIdx.x;
  if (i < n) {
    float d = deg[i];
    deg[i] = (d > 0.0f) ? rsqrtf(d) : 0.0f;
  }
}

// h = x @ W using V_WMMA_F32_16X16X4_F32.
// One wave computes a 16x64 tile of h. W (64 KB) is staged in LDS.
__global__ __launch_bounds__(256) void k_gemm_wmma(
    const float* __restrict__ x, const float* __restrict__ W,
    float* __restrict__ h, int n_rows) {
  __shared__ float Wl[IN_C * OUT_C];  // 64 KB of 320 KB/WGP
  // Cooperative float4 load of W into LDS.
  {
    const float4* Wv = (const float4*)W;
    float4* Lv = (float4*)Wl;
    for (int i = threadIdx.x; i < (IN_C * OUT_C) / 4; i += 256)
      Lv[i] = Wv[i];
  }
  __syncthreads();

  const int lane = threadIdx.x & 31;
  const int half = lane >> 4;   // 0: lanes 0-15, 1: lanes 16-31
  const int l16  = lane & 15;
  const int tile = blockIdx.x * 8 + (threadIdx.x >> 5);  // 16-row tile
  const int m0   = tile * 16;
  if (m0 < n_rows) {
    v8f c0 = {}, c1 = {}, c2 = {}, c3 = {};
    // A-frag: lane l16 owns row m0+l16; half selects K pair {k,k+1} vs {k+2,k+3}
    const float* xrow = x + (size_t)(m0 + l16) * IN_C + 2 * half;

    #pragma unroll 4
    for (int k = 0; k < IN_C; k += 4) {
      v2f a = *(const v2f*)(xrow + k);
      // B-frag (K x N layout): VGPR0 -> K rows {k, k+2}, VGPR1 -> {k+1, k+3}
      const float* wr = Wl + (size_t)(k + 2 * half) * OUT_C;
      v2f b0, b1, b2, b3;
      b0.x = wr[l16];       b0.y = wr[OUT_C + l16];
      b1.x = wr[16 + l16];  b1.y = wr[OUT_C + 16 + l16];
      b2.x = wr[32 + l16];  b2.y = wr[OUT_C + 32 + l16];
      b3.x = wr[48 + l16];  b3.y = wr[OUT_C + 48 + l16];
      c0 = __builtin_amdgcn_wmma_f32_16x16x4_f32(false, a, false, b0, (short)0, c0, false, false);
      c1 = __builtin_amdgcn_wmma_f32_16x16x4_f32(false, a, false, b1, (short)0, c1, false, false);
      c2 = __builtin_amdgcn_wmma_f32_16x16x4_f32(false, a, false, b2, (short)0, c2, false, false);
      c3 = __builtin_amdgcn_wmma_f32_16x16x4_f32(false, a, false, b3, (short)0, c3, false, false);
    }

    // C/D layout: VGPR v, lanes 0-15 -> row m0+v, lanes 16-31 -> row m0+v+8;
    // col = 16*j + l16.
    float* hrow = h + (size_t)(m0 + 8 * half) * OUT_C + l16;
    #pragma unroll
    for (int v = 0; v < 8; ++v) {
      hrow[(size_t)v * OUT_C +  0] = c0[v];
      hrow[(size_t)v * OUT_C + 16] = c1[v];
      hrow[(size_t)v * OUT_C + 32] = c2[v];
      hrow[(size_t)v * OUT_C + 48] = c3[v];
    }
  }
}

// out[i][c] = b[c] + dinv[i]^2 * h[i][c]   (bias + self-loop message)
__global__ void k_selfloop_bias(const float* __restrict__ dinv,
                                const float* __restrict__ h,
                                const float* __restrict__ bias,
                                float* __restrict__ out, int n) {
  int total = n * OUT_C;
  int stride = gridDim.x * blockDim.x;
  for (int i = blockIdx.x * blockDim.x + threadIdx.x; i < total; i += stride) {
    int node = i >> 6;
    int c = i & 63;
    float di = dinv[node];
    out[i] = bias[c] + di * di * h[i];
  }
}

// One wave per edge (grid-stride): gather h[row], scale by norm, atomic-add
// into out[col]. h and out are L2-resident (25.6 MB each), so this runs at
// L2 bandwidth. Each lane handles a contiguous float2 of the 64 channels.
__global__ __launch_bounds__(256) void k_scatter(
    const int* __restrict__ ei, const float* __restrict__ ew,
    const float* __restrict__ dinv, const float* __restrict__ h,
    float* __restrict__ out, int n_edges) {
  const int lane = threadIdx.x & 31;
  int wave   = (blockIdx.x * blockDim.x + threadIdx.x) >> 5;
  int nwaves = (gridDim.x * blockDim.x) >> 5;
  for (int e = wave; e < n_edges; e += nwaves) {
    int r = ei[e];
    int c = ei[n_edges + e];
    float nrm = dinv[r] * ew[e] * dinv[c];
    const float2 hv = *(const float2*)(h + (size_t)r * OUT_C + lane * 2);
    float* op = out + (size_t)c * OUT_C + lane * 2;
    atomAddF32(op,     nrm * hv.x);
    atomAddF32(op + 1, nrm * hv.y);
  }
}

extern "C" void kernel_launch(void* const* d_in, const int* in_sizes, int n_in,
                              void* d_out, int out_size, void* d_ws, size_t ws_size,
                              hipStream_t stream) {
  const float* x  = (const float*)d_in[0];
  const int*   ei = (const int*)d_in[1];   // [2, E] int32 (JAX default demotes int64)
  const float* ew = (const float*)d_in[2];
  const float* W  = (const float*)d_in[3];
  const float* b  = (const float*)d_in[4];
  float* out = (float*)d_out;

  const int n  = in_sizes[0] / IN_C;  // 100000
  const int ne = in_sizes[2];         // 1600000

  float* ws   = (float*)d_ws;
  float* dinv = ws;                                  // N floats (deg -> dinv)
  size_t hoff = (((size_t)n + 63) / 64) * 64;        // 256B-align h
  float* h    = ws + hoff;                           // N*64 floats

  k_init_deg<<<(n + 255) / 256, 256, 0, stream>>>(dinv, n);
  k_deg<<<1024, 256, 0, stream>>>(ei, ew, dinv, ne);
  k_dinv<<<(n + 255) / 256, 256, 0, stream>>>(dinv, n);

  int tiles = (n + 15) / 16;                         // 6250 (exact)
  k_gemm_wmma<<<(tiles + 7) / 8, 256, 0, stream>>>(x, W, h, n);

  k_selfloop_bias<<<4096, 256, 0, stream>>>(dinv, h, b, out, n);
  k_scatter<<<2048, 256, 0, stream>>>(ei, ew, dinv, h, out, ne);
}